// ConditionNumberLoss_9758165696607
// MI455X (gfx1250) — compile-verified
//
#include <hip/hip_runtime.h>
#include <math.h>

#define NPTS 8192
#define NB   2
#define TK   16
#define BALL2 0.2f

typedef __attribute__((ext_vector_type(2))) float v2f;
typedef __attribute__((ext_vector_type(8))) float v8f;

// ---------------------------------------------------------------------------
// Kernel 1: KNN via V_WMMA_F32_16X16X4_F32.
// One wave32 handles 16 query rows; loops over db in 16-column tiles.
// Distance tile computed on the matrix core in full fp32:
//   A row m = [q.x q.y q.z 1],  B col n = [-2p.x -2p.y -2p.z ||p||^2]
//   D = -2 q.p + ||p||^2 ;  + ||q||^2 added per-row during selection.
// ---------------------------------------------------------------------------
__global__ __launch_bounds__(256) void knn_kernel(const float* __restrict__ ref,
                                                  float* __restrict__ odist,
                                                  int* __restrict__ oidx)
{
    // [wave][row m][col n], stride 20 floats: rows 16B-aligned, stores from the
    // two lane-halves hit disjoint bank sets (20r+lm vs 20r+lm+32), b128 reads
    // at spacing-20 ranges are conflict-free.
    __shared__ float tile[8][16][20];

    const int lane = threadIdx.x & 31;
    const int wave = threadIdx.x >> 5;
    const int rb   = blockIdx.x * 8 + wave;        // 0..1023 row-blocks
    const int b    = rb >> 9;                      // / (NPTS/16)
    const int qbase = (rb & 511) << 4;
    const float* refb = ref + (size_t)b * NPTS * 3;

    const int  lm = lane & 15;
    const bool hi = lane >= 16;
    const int  moff = hi ? 8 : 0;

    // Query point for this lane's assigned row (rows duplicated across halves).
    float qx = refb[(qbase + lm) * 3 + 0];
    float qy = refb[(qbase + lm) * 3 + 1];
    float qz = refb[(qbase + lm) * 3 + 2];
    float qn = qx * qx + qy * qy + qz * qz;

    // A-matrix 16x4 f32 layout: lanes 0-15 -> {K0,K1}, lanes 16-31 -> {K2,K3}.
    v2f a;
    a.x = hi ? qz : qx;
    a.y = hi ? 1.0f : qy;

    float bd[TK];
    int   bi[TK];
#pragma unroll
    for (int k = 0; k < TK; ++k) { bd[k] = 3.0e38f; bi[k] = 0; }

    for (int t = 0; t < NPTS; t += 16) {
        // db point for this lane's column (both halves load the same point).
        float px = refb[(t + lm) * 3 + 0];
        float py = refb[(t + lm) * 3 + 1];
        float pz = refb[(t + lm) * 3 + 2];
        if (t + 16 < NPTS)   // hint next tile (global_prefetch_b8)
            __builtin_prefetch(&refb[(t + 16 + lm) * 3], 0, 3);

        v2f bb;
        bb.x = hi ? (-2.0f * pz) : (-2.0f * px);
        bb.y = hi ? (px * px + py * py + pz * pz) : (-2.0f * py);

        v8f acc = {};
        acc = __builtin_amdgcn_wmma_f32_16x16x4_f32(
            false, a, false, bb, (short)0, acc, false, false);

        // D layout: VGPR r holds row m = r + (lane<16 ? 0 : 8), col = lane&15.
        // Scatter into row-major tile: 8 strided b32 stores, conflict-free.
#pragma unroll
        for (int r = 0; r < 8; ++r)
            tile[wave][r + moff][lm] = acc[r];

        asm volatile("s_wait_dscnt 0" ::: "memory");

        // Each lane owns row m = lane&15: 4 independent b128 loads pull the
        // whole candidate row into registers before any compare.
        const float4* rowp = (const float4*)&tile[wave][lm][0];
        float4 q0 = rowp[0], q1 = rowp[1], q2 = rowp[2], q3 = rowp[3];
        float dv[16];
        dv[0]=q0.x+qn;  dv[1]=q0.y+qn;  dv[2]=q0.z+qn;  dv[3]=q0.w+qn;
        dv[4]=q1.x+qn;  dv[5]=q1.y+qn;  dv[6]=q1.z+qn;  dv[7]=q1.w+qn;
        dv[8]=q2.x+qn;  dv[9]=q2.y+qn;  dv[10]=q2.z+qn; dv[11]=q2.w+qn;
        dv[12]=q3.x+qn; dv[13]=q3.y+qn; dv[14]=q3.z+qn; dv[15]=q3.w+qn;

        float tmin = dv[0];
#pragma unroll
        for (int n = 1; n < 16; ++n) tmin = fminf(tmin, dv[n]);

        float worst = bd[TK - 1];
        if (tmin < worst) {
#pragma unroll
            for (int n = 0; n < 16; ++n) {
                float d = dv[n];
                if (d < worst) {
                    bd[TK - 1] = d;
                    bi[TK - 1] = t + n;
#pragma unroll
                    for (int j = TK - 1; j > 0; --j) {
                        if (bd[j] < bd[j - 1]) {
                            float td = bd[j]; bd[j] = bd[j - 1]; bd[j - 1] = td;
                            int   ti = bi[j]; bi[j] = bi[j - 1]; bi[j - 1] = ti;
                        }
                    }
                    worst = bd[TK - 1];
                }
            }
        }
        asm volatile("s_wait_dscnt 0" ::: "memory");
    }

    if (lane < 16) {
        size_t base = ((size_t)(b * NPTS + qbase + lm)) * TK;
#pragma unroll
        for (int k = 0; k < TK; ++k) {
            odist[base + k] = bd[k];
            oidx[base + k]  = bi[k];
        }
    }
}

// ---------------------------------------------------------------------------
// Closed-form symmetric 3x3 eigenvalues -> cond = smax / (smin + smax)
// ---------------------------------------------------------------------------
__device__ __forceinline__ float cond_from_gram(float g00, float g01, float g02,
                                                float g11, float g12, float g22)
{
    float q  = (g00 + g11 + g22) * (1.0f / 3.0f);
    float b00 = g00 - q, b11 = g11 - q, b22 = g22 - q;
    float p2 = b00 * b00 + b11 * b11 + b22 * b22
             + 2.0f * (g01 * g01 + g02 * g02 + g12 * g12);
    float p   = sqrtf(p2 * (1.0f / 6.0f));
    float inv = (p > 0.0f) ? (1.0f / p) : 0.0f;
    float c00 = b00 * inv, c01 = g01 * inv, c02 = g02 * inv;
    float c11 = b11 * inv, c12 = g12 * inv, c22 = b22 * inv;
    float detB = c00 * (c11 * c22 - c12 * c12)
               - c01 * (c01 * c22 - c12 * c02)
               + c02 * (c01 * c12 - c11 * c02);
    float r = fminf(fmaxf(0.5f * detB, -1.0f), 1.0f);
    float phi = acosf(r) * (1.0f / 3.0f);
    float e1 = q + 2.0f * p * __cosf(phi);                   // lambda max
    float e3 = q + 2.0f * p * __cosf(phi + 2.09439510239f);  // lambda min
    float s0 = sqrtf(fmaxf(e1, 0.0f));
    float s2 = sqrtf(fmaxf(e3, 0.0f));
    return s0 / (s2 + s0);
}

// ---------------------------------------------------------------------------
// Kernel 2: per-point condition numbers (ref masked, points unmasked — the
// reference discards the masked_fill on `points`), squared diff to ws.
// ---------------------------------------------------------------------------
__global__ __launch_bounds__(256) void cond_kernel(const float* __restrict__ ref,
                                                   const float* __restrict__ pts,
                                                   const float* __restrict__ dist,
                                                   const int* __restrict__ idx,
                                                   float* __restrict__ sq)
{
    int t = blockIdx.x * 256 + threadIdx.x;
    if (t >= NB * NPTS) return;
    int b = t >> 13;
    const float* refb = ref + (size_t)b * NPTS * 3;
    const float* ptsb = pts + (size_t)b * NPTS * 3;
    size_t base = (size_t)t * TK;

    float nball = 0.0f;
    float r1x = 0, r1y = 0, r1z = 0;
    float rxx = 0, rxy = 0, rxz = 0, ryy = 0, ryz = 0, rzz = 0;
    float p1x = 0, p1y = 0, p1z = 0;
    float pxx = 0, pxy = 0, pxz = 0, pyy = 0, pyz = 0, pzz = 0;

#pragma unroll
    for (int k = 0; k < TK; ++k) {
        float dk = dist[base + k];
        int   j  = idx[base + k];
        bool  m  = dk < BALL2;
        nball += m ? 1.0f : 0.0f;

        float rx = refb[j * 3 + 0], ry = refb[j * 3 + 1], rz = refb[j * 3 + 2];
        rx = m ? rx : 0.0f; ry = m ? ry : 0.0f; rz = m ? rz : 0.0f;
        r1x += rx; r1y += ry; r1z += rz;
        rxx += rx * rx; rxy += rx * ry; rxz += rx * rz;
        ryy += ry * ry; ryz += ry * rz; rzz += rz * rz;

        float px = ptsb[j * 3 + 0], py = ptsb[j * 3 + 1], pz = ptsb[j * 3 + 2];
        p1x += px; p1y += py; p1z += pz;
        pxx += px * px; pxy += px * py; pxz += px * pz;
        pyy += py * py; pyz += py * pz; pzz += pz * pz;
    }

    float invn = 1.0f / nball;  // self-distance 0 < BALL2 -> nball >= 1
    // Gram of centered group: S2 - c*S1^T - S1*c^T + K*c*c^T, c = S1/nball
    float cx = r1x * invn, cy = r1y * invn, cz = r1z * invn;
    float g00 = rxx - 2.0f * cx * r1x + (float)TK * cx * cx;
    float g01 = rxy - cx * r1y - cy * r1x + (float)TK * cx * cy;
    float g02 = rxz - cx * r1z - cz * r1x + (float)TK * cx * cz;
    float g11 = ryy - 2.0f * cy * r1y + (float)TK * cy * cy;
    float g12 = ryz - cy * r1z - cz * r1y + (float)TK * cy * cz;
    float g22 = rzz - 2.0f * cz * r1z + (float)TK * cz * cz;
    float ref_cond = cond_from_gram(g00, g01, g02, g11, g12, g22);

    cx = p1x * invn; cy = p1y * invn; cz = p1z * invn;
    g00 = pxx - 2.0f * cx * p1x + (float)TK * cx * cx;
    g01 = pxy - cx * p1y - cy * p1x + (float)TK * cx * cy;
    g02 = pxz - cx * p1z - cz * p1x + (float)TK * cx * cz;
    g11 = pyy - 2.0f * cy * p1y + (float)TK * cy * cy;
    g12 = pyz - cy * p1z - cz * p1y + (float)TK * cy * cz;
    g22 = pzz - 2.0f * cz * p1z + (float)TK * cz * cz;
    float cond = cond_from_gram(g00, g01, g02, g11, g12, g22);

    float diff = cond - ref_cond;
    sq[t] = diff * diff;
}

// ---------------------------------------------------------------------------
// Kernel 3: deterministic single-block tree reduction -> mean
// ---------------------------------------------------------------------------
__global__ __launch_bounds__(256) void reduce_kernel(const float* __restrict__ sq,
                                                     float* __restrict__ out)
{
    __shared__ float red[256];
    float s = 0.0f;
    for (int i = threadIdx.x; i < NB * NPTS; i += 256) s += sq[i];
    red[threadIdx.x] = s;
    __syncthreads();
    for (int off = 128; off > 0; off >>= 1) {
        if ((int)threadIdx.x < off) red[threadIdx.x] += red[threadIdx.x + off];
        __syncthreads();
    }
    if (threadIdx.x == 0) out[0] = red[0] * (1.0f / (float)(NB * NPTS));
}

// ---------------------------------------------------------------------------
extern "C" void kernel_launch(void* const* d_in, const int* in_sizes, int n_in,
                              void* d_out, int out_size, void* d_ws, size_t ws_size,
                              hipStream_t stream)
{
    const float* ref = (const float*)d_in[0];   // ref_points (2,8192,3)
    const float* pts = (const float*)d_in[1];   // points     (2,8192,3)

    float* ws   = (float*)d_ws;
    float* dist = ws;                               // B*N*K = 262144 floats
    int*   idx  = (int*)(ws + NB * NPTS * TK);      // 262144 ints
    float* sq   = ws + 2 * NB * NPTS * TK;          // B*N = 16384 floats

    knn_kernel<<<128, 256, 0, stream>>>(ref, dist, idx);       // 1024 waves
    cond_kernel<<<(NB * NPTS + 255) / 256, 256, 0, stream>>>(ref, pts, dist, idx, sq);
    reduce_kernel<<<1, 256, 0, stream>>>(sq, (float*)d_out);
}